// YoloLoss_12945031430825
// MI455X (gfx1250) — compile-verified
//
#include <hip/hip_runtime.h>
#include <hip/hip_bf16.h>
#include <math.h>

// ---------------------------------------------------------------------------
// YOLOv5-style loss on MI455X (gfx1250, wave32).
// Structure: tiny scatter/gather kernels + big elementwise BCE reductions.
// Wave-level reductions use V_WMMA_F32_16X16X4_F32 with a ones B matrix:
//   A[m,0] = partial of lane m, A[m,2] = partial of lane m+16, A[*,1]=A[*,3]=0
//   B = all ones  ->  D[m,n] = p_m + p_{m+16}
//   lane sums its 8 accumulator VGPRs, shfl_xor(16) -> full 32-lane total.
// ---------------------------------------------------------------------------

typedef __attribute__((ext_vector_type(2))) float v2f;
typedef __attribute__((ext_vector_type(8))) float v8f;

#define NCLS 80
#define NT   512
#define NB   16

__device__ __forceinline__ float bce_logits(float x, float z) {
    return fmaxf(x, 0.0f) - x * z + log1pf(expf(-fabsf(x)));
}

__device__ __forceinline__ float sigm(float x) {
    return 1.0f / (1.0f + expf(-x));
}

// Sum of all 32 lanes' v, returned to every lane. Requires full wave (EXEC=~0).
__device__ __forceinline__ float wave_sum_wmma(float v) {
    v2f a; a[0] = v;     a[1] = 0.0f;
    v2f b; b[0] = 1.0f;  b[1] = 1.0f;
    v8f c = {};
    c = __builtin_amdgcn_wmma_f32_16x16x4_f32(false, a, false, b,
                                              (short)0, c, false, false);
    float s = c[0] + c[1] + c[2] + c[3] + c[4] + c[5] + c[6] + c[7];
    s += __shfl_xor(s, 16, 32);   // combine M=0..7 half with M=8..15 half
    return s;
}

__device__ __forceinline__ float giou_fn(float ax, float ay, float aw, float ah,
                                         float bx, float by, float bw, float bh) {
    float a_x0 = ax - aw * 0.5f, a_x1 = ax + aw * 0.5f;
    float a_y0 = ay - ah * 0.5f, a_y1 = ay + ah * 0.5f;
    float b_x0 = bx - bw * 0.5f, b_x1 = bx + bw * 0.5f;
    float b_y0 = by - bh * 0.5f, b_y1 = by + bh * 0.5f;
    float iw = fmaxf(fminf(a_x1, b_x1) - fmaxf(a_x0, b_x0), 0.0f);
    float ih = fmaxf(fminf(a_y1, b_y1) - fmaxf(a_y0, b_y0), 0.0f);
    float inter = iw * ih;
    float uni = (a_x1 - a_x0) * (a_y1 - a_y0) +
                (b_x1 - b_x0) * (b_y1 - b_y0) - inter;
    float iou = inter / uni;
    float cw = fmaxf(a_x1, b_x1) - fminf(a_x0, b_x0) + 1e-16f;
    float ch = fmaxf(a_y1, b_y1) - fminf(a_y0, b_y0);
    float carea = cw * ch + 1e-16f;
    return iou - (carea - uni) / carea;
}

__global__ void zero_ws(float* __restrict__ ws, int n) {
    for (int i = blockIdx.x * blockDim.x + threadIdx.x; i < n;
         i += gridDim.x * blockDim.x)
        ws[i] = 0.0f;
}

// One thread per (offset o, anchor a, target n) candidate; grid is exactly
// 7680 threads (30 blocks x 256) so every wave is full -> WMMA-legal.
__global__ void cand_kernel(const float* __restrict__ layer,
                            const float* __restrict__ tg,
                            float* __restrict__ objgt,
                            float* __restrict__ acc,
                            int W, int H, int lvl,
                            float a0w, float a0h, float a1w, float a1h,
                            float a2w, float a2h) {
    const int e = blockIdx.x * blockDim.x + threadIdx.x;   // 0..7679
    const int n = e % NT;
    const int a = (e / NT) % 3;
    const int o = e / (NT * 3);
    const int HW = W * H;

    float img = tg[n * 6 + 0];
    float cls = tg[n * 6 + 1];
    float tx  = tg[n * 6 + 2] * (float)W;
    float ty  = tg[n * 6 + 3] * (float)H;
    float tw  = tg[n * 6 + 4] * (float)W;
    float th  = tg[n * 6 + 5] * (float)H;

    float aw = (a == 0) ? a0w : ((a == 1) ? a1w : a2w);
    float ah = (a == 0) ? a0h : ((a == 1) ? a1h : a2h);

    // anchor ratio selection
    float rw = tw / aw, rh = th / ah;
    float mr = fmaxf(fmaxf(rw, 1.0f / rw), fmaxf(rh, 1.0f / rh));
    bool sel = mr < 4.0f;

    // offset-condition selection (uniform per block: o,a constant over 256 thr)
    float fx = tx - floorf(tx);
    float fy = ty - floorf(ty);
    bool cond; float offx = 0.0f, offy = 0.0f;
    switch (o) {
        case 0:  cond = true; break;
        case 1:  cond = (fx < 0.5f) && (tx > 1.0f);              offx =  0.5f; break;
        case 2:  cond = (fy < 0.5f) && (ty > 1.0f);              offy =  0.5f; break;
        case 3:  cond = (fx > 0.5f) && (tx < (float)W - 1.0f);   offx = -0.5f; break;
        default: cond = (fy > 0.5f) && (ty < (float)H - 1.0f);   offy = -0.5f; break;
    }
    bool valid = cond && sel;
    float vf = valid ? 1.0f : 0.0f;

    int gridx = (int)(tx - offx);      // trunc, matches astype(int32)
    int gridy = (int)(ty - offy);
    float gxf = tx - (float)gridx;     // gxy uses UNclipped grid
    float gyf = ty - (float)gridy;
    int gx = min(max(gridx, 0), W - 1);
    int gy = min(max(gridy, 0), H - 1);
    int im = (int)img;
    int ci = (int)cls;

    // pred[b,a,h,w,c] = layer[((b*255)+(a*85+c))*HW + h*W + w]
    long long base = ((long long)im * 255 + a * 85) * (long long)HW
                     + (long long)gy * W + gx;

    float x0 = layer[base + 0 * (long long)HW];
    float x1 = layer[base + 1 * (long long)HW];
    float x2 = layer[base + 2 * (long long)HW];
    float x3 = layer[base + 3 * (long long)HW];

    float px = sigm(x0) * 2.0f - 0.5f;
    float py = sigm(x1) * 2.0f - 0.5f;
    float sw = sigm(x2) * 2.0f; float pw = sw * sw * aw;
    float sh = sigm(x3) * 2.0f; float ph = sh * sh * ah;

    float giou = giou_fn(px, py, pw, ph, gxf, gyf, tw, th);
    float box_term = (1.0f - giou) * vf;

    float csum = 0.0f;
    #pragma unroll 4
    for (int c = 0; c < NCLS; ++c) {
        float xc = layer[base + (long long)(5 + c) * HW];
        float z = (c == ci) ? 1.0f : 0.0f;
        csum += bce_logits(xc, z);
    }
    csum *= vf;

    // scatter giou_t into obj ground-truth map (set semantics; dup order
    // unspecified in the reference too)
    if (valid) {
        objgt[((long long)im * 3 + a) * (long long)HW + (long long)gy * W + gx]
            = fmaxf(giou, 0.0f);
    }

    // wave reductions via WMMA, one atomic per wave per accumulator
    float bs = wave_sum_wmma(box_term);
    float cs = wave_sum_wmma(csum);
    float ns = wave_sum_wmma(vf);
    if ((threadIdx.x & 31) == 0) {
        atomicAdd(&acc[0 + lvl], bs);
        atomicAdd(&acc[3 + lvl], cs);
        atomicAdd(&acc[6 + lvl], ns);
    }
}

// BCE over the full (B,3,H,W) objectness map vs scattered obj_gt.
__global__ void obj_kernel(const float* __restrict__ layer,
                           const float* __restrict__ objgt,
                           float* __restrict__ acc_obj,
                           int H, int W) {
    const int HW = H * W;
    const int total = NB * 3 * HW;
    float local = 0.0f;
    for (int i = blockIdx.x * blockDim.x + threadIdx.x; i < total;
         i += gridDim.x * blockDim.x) {
        int b  = i / (3 * HW);
        int r  = i - b * 3 * HW;
        int a  = r / HW;
        int hw = r - a * HW;
        float x = layer[((long long)b * 255 + a * 85 + 4) * (long long)HW + hw];
        local += bce_logits(x, objgt[i]);
    }
    float s = wave_sum_wmma(local);
    if ((threadIdx.x & 31) == 0) atomicAdd(acc_obj, s);
}

__global__ void combine_kernel(const float* __restrict__ acc,
                               float* __restrict__ out) {
    const float bal[3] = {4.0f, 1.0f, 0.4f};
    const int   hw[3]  = {80 * 80, 40 * 40, 20 * 20};
    float lbox = 0.0f, lobj = 0.0f, lcls = 0.0f;
    for (int l = 0; l < 3; ++l) {
        float cnt   = acc[6 + l];
        float denom = fmaxf(cnt, 1.0f);
        float box = (cnt > 0.0f) ? acc[0 + l] / denom : 0.0f;
        float cls = (cnt > 0.0f) ? acc[3 + l] / (denom * (float)NCLS) : 0.0f;
        float obj = acc[9 + l] / (float)(NB * 3 * hw[l]);
        lbox += box;
        lcls += cls;
        lobj += obj * bal[l];
    }
    // BOX_W=0.05, OBJ_W=1.0, CLS_W=0.5*NC/80=0.5, scale=1, *batch_size
    out[0] = (lbox * 0.05f + lobj * 1.0f + lcls * 0.5f) * (float)NB;
}

extern "C" void kernel_launch(void* const* d_in, const int* in_sizes, int n_in,
                              void* d_out, int out_size, void* d_ws, size_t ws_size,
                              hipStream_t stream) {
    const float* p0 = (const float*)d_in[0];   // (16,255,80,80)
    const float* p1 = (const float*)d_in[1];   // (16,255,40,40)
    const float* p2 = (const float*)d_in[2];   // (16,255,20,20)
    const float* tg = (const float*)d_in[3];   // (512,6)
    float* out = (float*)d_out;
    float* ws  = (float*)d_ws;

    // workspace layout: [0..15] accumulators (box[3], cls[3], cnt[3], obj[3]),
    // then obj_gt maps for levels 0,1,2.
    float* acc = ws;
    float* og0 = ws + 16;
    float* og1 = og0 + (long long)NB * 3 * 80 * 80;
    float* og2 = og1 + (long long)NB * 3 * 40 * 40;
    const int nzero = 16 + NB * 3 * (80 * 80 + 40 * 40 + 20 * 20);

    zero_ws<<<256, 256, 0, stream>>>(ws, nzero);

    // 5 offsets * 3 anchors * 512 targets = 7680 threads, exact grid
    cand_kernel<<<30, 256, 0, stream>>>(p0, tg, og0, acc, 80, 80, 0,
                                        10.f, 13.f, 16.f, 30.f, 33.f, 23.f);
    cand_kernel<<<30, 256, 0, stream>>>(p1, tg, og1, acc, 40, 40, 1,
                                        30.f, 61.f, 62.f, 45.f, 59.f, 119.f);
    cand_kernel<<<30, 256, 0, stream>>>(p2, tg, og2, acc, 20, 20, 2,
                                        116.f, 90.f, 156.f, 198.f, 373.f, 326.f);

    obj_kernel<<<512, 256, 0, stream>>>(p0, og0, &acc[9],  80, 80);
    obj_kernel<<<256, 256, 0, stream>>>(p1, og1, &acc[10], 40, 40);
    obj_kernel<<<128, 256, 0, stream>>>(p2, og2, &acc[11], 20, 20);

    combine_kernel<<<1, 1, 0, stream>>>(acc, out);
}